// SpatialAttentionModule_22351009808867
// MI455X (gfx1250) — compile-verified
//
#include <hip/hip_runtime.h>
#include <cstdint>

// ---- problem geometry (fixed by the reference) ----
#define B_   16
#define C_   64
#define H_   256
#define W_   256
#define HW_  (H_ * W_)
#define KS_  7
#define PAD_ 3

typedef float f4 __attribute__((ext_vector_type(4)));

// ---- address-space helpers (CK-style integer-cast to AS1/AS3) ----
// The gfx1250 async builtin takes AS-qualified int* params.
typedef __attribute__((address_space(3))) int lds_int_t;
typedef __attribute__((address_space(1))) int gbl_int_t;

static __device__ __forceinline__ lds_int_t* as_lds(const void* p) {
    return (lds_int_t*)(uint32_t)(uintptr_t)p;   // low 32 bits == LDS byte offset
}
static __device__ __forceinline__ gbl_int_t* as_gbl(const void* p) {
    return (gbl_int_t*)(uintptr_t)p;
}

// =====================================================================
// Phase 1: per-pixel channel mean + max  ->  mask[b][0..1][h][w] (in ws)
// Each thread owns one float4 (4 pixels), loops over 64 channels.
// Coalesced b128 streaming; ascending order (tail of x stays in L2).
// =====================================================================
__global__ __launch_bounds__(256) void sam_reduce(const float* __restrict__ x,
                                                  float* __restrict__ mask)
{
    const int tid = threadIdx.x;
    const int blocksPerBatch = HW_ / (256 * 4);          // 64
    const int b   = blockIdx.x / blocksPerBatch;
    const int blk = blockIdx.x % blocksPerBatch;
    const int q   = blk * 256 + tid;                     // float4 index in plane
    const int planeq = HW_ / 4;                          // 16384

    const f4* xb = (const f4*)(x + (size_t)b * C_ * HW_);
    f4 v = xb[q];                                        // c = 0
    f4 s = v;
    f4 m = v;
#pragma unroll 8
    for (int c = 1; c < C_; ++c) {
        f4 t = xb[(size_t)c * planeq + q];
        s += t;
        m.x = fmaxf(m.x, t.x); m.y = fmaxf(m.y, t.y);
        m.z = fmaxf(m.z, t.z); m.w = fmaxf(m.w, t.w);
    }
    const float inv = 1.0f / (float)C_;
    f4 a = s * inv;
    f4* mk = (f4*)(mask + (size_t)b * 2 * HW_);
    mk[q]          = a;                                  // channel 0: mean
    mk[planeq + q] = m;                                  // channel 1: max
}

// =====================================================================
// Phase 2 (fused): 7x7 conv(2->1) on mask + sigmoid + x * att broadcast.
// 64x16-pixel tile per 256-thread block; halo (70x22 x 2ch) pulled into
// LDS via CDNA5 async-to-LDS; blocks walked in REVERSE so the L2-resident
// tail of x from phase 1 is consumed first.
// =====================================================================
#define TW 64
#define TH 16
#define HALO_W  (TW + 2 * PAD_)   // 70
#define HALO_H  (TH + 2 * PAD_)   // 22
#define HALO_WP 72                // padded row stride (floats)

__global__ __launch_bounds__(256) void sam_conv_mul(const float* __restrict__ x,
                                                    const float* __restrict__ mask,
                                                    const float* __restrict__ w,
                                                    const float* __restrict__ bias,
                                                    float* __restrict__ out)
{
    __shared__ float sm[2][HALO_H][HALO_WP];             // ~12.7 KB
    __shared__ float swt[2 * KS_ * KS_];

    const int tid = threadIdx.x;

    const int tilesX = W_ / TW;                          // 4
    const int tilesY = H_ / TH;                          // 16
    const int nTiles = B_ * tilesX * tilesY;             // 1024
    const int id  = (nTiles - 1) - (int)blockIdx.x;      // reversed order
    const int b   = id / (tilesX * tilesY);
    const int t2  = id % (tilesX * tilesY);
    const int ty0 = (t2 / tilesX) * TH;
    const int tx0 = (t2 % tilesX) * TW;

    if (tid < 2 * KS_ * KS_) swt[tid] = w[tid];

    // zero-fill halo only on image-border tiles (zero 'same' padding)
    const bool edge = (tx0 == 0) | (ty0 == 0) | (tx0 + TW == W_) | (ty0 + TH == H_);
    if (edge) {
        for (int i = tid; i < 2 * HALO_H * HALO_WP; i += 256)
            ((float*)sm)[i] = 0.0f;
    }
    __syncthreads();

    // ---- async halo fill: global -> LDS without touching VGPRs ----
#pragma unroll
    for (int ch = 0; ch < 2; ++ch) {
        const float* mp = mask + ((size_t)b * 2 + ch) * HW_;
        for (int i = tid; i < HALO_H * HALO_W; i += 256) {
            const int r  = i / HALO_W;
            const int cc = i - r * HALO_W;
            const int gy = ty0 - PAD_ + r;
            const int gx = tx0 - PAD_ + cc;
            if ((unsigned)gy < (unsigned)H_ && (unsigned)gx < (unsigned)W_) {
                __builtin_amdgcn_global_load_async_to_lds_b32(
                    as_gbl(mp + gy * W_ + gx), as_lds(&sm[ch][r][cc]),
                    /*imm offset*/0, /*cpol*/0);
            }
        }
    }

    // Thread's pixel-quad coordinates + x/out base (needed for prefetch now)
    const int txq = tid & 15;        // float4 column within tile
    const int ry  = tid >> 4;        // row within tile
    const int px  = txq * 4;         // pixel column within tile
    const size_t pq = (((size_t)b * C_) * HW_
                     + (size_t)(ty0 + ry) * W_ + (size_t)(tx0 + px)) >> 2;
    const f4* xp = (const f4*)x + pq;
    f4*       op = (f4*)out + pq;

    // Warm HBM/L2 for the multiply phase while we do the conv (CDNA5
    // global_prefetch); one line per of the first 4 channel planes.
#pragma unroll
    for (int c = 0; c < 4; ++c)
        __builtin_prefetch(xp + (size_t)c * (HW_ / 4), 0, 0);

    asm volatile("s_wait_asynccnt 0" ::: "memory");
    __syncthreads();

    // ---- conv 7x7 (2 in-ch) + bias + sigmoid for this thread's 4 pixels ----
    const float bv = bias[0];
    float a0 = bv, a1 = bv, a2 = bv, a3 = bv;
#pragma unroll
    for (int ch = 0; ch < 2; ++ch) {
#pragma unroll
        for (int ky = 0; ky < KS_; ++ky) {
            const float* row = &sm[ch][ry + ky][px];
            float seg[10];
#pragma unroll
            for (int t = 0; t < 10; ++t) seg[t] = row[t];
#pragma unroll
            for (int kx = 0; kx < KS_; ++kx) {
                const float wv = swt[ch * 49 + ky * 7 + kx];
                a0 = fmaf(seg[kx + 0], wv, a0);
                a1 = fmaf(seg[kx + 1], wv, a1);
                a2 = fmaf(seg[kx + 2], wv, a2);
                a3 = fmaf(seg[kx + 3], wv, a3);
            }
        }
    }
    f4 att;
    att.x = 1.0f / (1.0f + expf(-a0));
    att.y = 1.0f / (1.0f + expf(-a1));
    att.z = 1.0f / (1.0f + expf(-a2));
    att.w = 1.0f / (1.0f + expf(-a3));

    // ---- broadcast multiply over 64 channels: out = x * att ----
    // NT hints: last use of x, streaming store of out (don't thrash L2).
    // Unroll 8 -> 8 independent b128 loads in flight per wave.
#pragma unroll 8
    for (int c = 0; c < C_; ++c) {
        f4 v = __builtin_nontemporal_load(xp + (size_t)c * (HW_ / 4));
        v *= att;
        __builtin_nontemporal_store(v, op + (size_t)c * (HW_ / 4));
    }
}

// =====================================================================
extern "C" void kernel_launch(void* const* d_in, const int* in_sizes, int n_in,
                              void* d_out, int out_size, void* d_ws, size_t ws_size,
                              hipStream_t stream)
{
    const float* x    = (const float*)d_in[0];   // (16,64,256,256)
    const float* w    = (const float*)d_in[1];   // (1,2,7,7)
    const float* bias = (const float*)d_in[2];   // (1,)
    float* out  = (float*)d_out;
    float* mask = (float*)d_ws;                  // 16*2*256*256 fp32 = 8 MB scratch

    sam_reduce  <<<dim3(B_ * (HW_ / (256 * 4))), dim3(256), 0, stream>>>(x, mask);
    sam_conv_mul<<<dim3(B_ * (W_ / TW) * (H_ / TH)), dim3(256), 0, stream>>>(x, mask, w, bias, out);
}